// PolynomialAttentionNet_88364657148295
// MI455X (gfx1250) — compile-verified
//
#include <hip/hip_runtime.h>

typedef float v2f __attribute__((ext_vector_type(2)));
typedef float v8f __attribute__((ext_vector_type(8)));

#define D_MODEL   256
#define INPUT_DIM 64
#define SEQ       4096
#define BATCH     4
#define RES_SCALE 0.1f
#define GSPLIT    8    // S-split chunks for Gram partials
#define CSPLIT    32   // S-split chunks for column-sum partials

// ---------------------------------------------------------------------------
// f32 WMMA: D(16x16) = A(16x4) * B(4x16) + C, wave32.
// ---------------------------------------------------------------------------
__device__ __forceinline__ v8f wmma_f32(v2f a, v2f b, v8f c) {
  return __builtin_amdgcn_wmma_f32_16x16x4_f32(false, a, false, b, (short)0, c,
                                               false, false);
}

// A fragment, row-major A[*, lda]: lanes 0-15 K=k0,k0+1 ; lanes 16-31 K=k0+2,k0+3
__device__ __forceinline__ v2f load_a_rm(const float* __restrict__ A, int lda,
                                         int m0, int k0, int lane) {
  int row = lane & 15;
  int kk  = k0 + ((lane >> 4) << 1);
  return *(const v2f*)(A + (size_t)(m0 + row) * lda + kk);
}

// A^T fragment: logical A(m,k) = stored[k, m]  (stored row-major [K, lda])
__device__ __forceinline__ v2f load_a_tr(const float* __restrict__ A, int lda,
                                         int m0, int k0, int lane) {
  int row = lane & 15;
  int kk  = k0 + ((lane >> 4) << 1);
  v2f a;
  a.x = A[(size_t)kk * lda + m0 + row];
  a.y = A[(size_t)(kk + 1) * lda + m0 + row];
  return a;
}

// B fragment, row-major B[K, ldb]
__device__ __forceinline__ v2f load_b_rm(const float* __restrict__ Bm, int ldb,
                                         int n0, int k0, int lane) {
  int col = lane & 15;
  int kk  = k0 + ((lane >> 4) << 1);
  v2f b;
  b.x = Bm[(size_t)kk * ldb + n0 + col];
  b.y = Bm[(size_t)(kk + 1) * ldb + n0 + col];
  return b;
}

// D/C layout: lane L, reg r -> row = m0 + (L>>4)*8 + r, col = n0 + (L&15)
__device__ __forceinline__ void store_d(float* __restrict__ Dm, int ldd, int m0,
                                        int n0, int lane, v8f acc) {
  int col   = n0 + (lane & 15);
  int rbase = m0 + ((lane >> 4) << 3);
#pragma unroll
  for (int r = 0; r < 8; ++r)
    Dm[(size_t)(rbase + r) * ldd + col] = acc[r];
}

// ---------------------------------------------------------------------------
// 1) h = x @ W_embed + b_embed   [B*S,64]@[64,256]; wave = 16 rows x 64 cols
// grid: (4, B*S/16), block: 32
// ---------------------------------------------------------------------------
__global__ void k_embed(const float* __restrict__ x, const float* __restrict__ We,
                        const float* __restrict__ be, float* __restrict__ h) {
  int lane = threadIdx.x & 31;
  int n0 = blockIdx.x * 64;
  int m0 = blockIdx.y * 16;
  v8f acc[4] = {};
#pragma unroll
  for (int k0 = 0; k0 < INPUT_DIM; k0 += 4) {
    v2f a = load_a_rm(x, INPUT_DIM, m0, k0, lane);   // 1 b64 load feeds 4 WMMAs
#pragma unroll
    for (int j = 0; j < 4; ++j) {
      v2f b = load_b_rm(We, D_MODEL, n0 + 16 * j, k0, lane);
      acc[j] = wmma_f32(a, b, acc[j]);
    }
  }
  int colb  = lane & 15;
  int rbase = m0 + ((lane >> 4) << 3);
#pragma unroll
  for (int j = 0; j < 4; ++j) {
    float bias = be[n0 + 16 * j + colb];
#pragma unroll
    for (int r = 0; r < 8; ++r)
      h[(size_t)(rbase + r) * D_MODEL + n0 + 16 * j + colb] = acc[j][r] + bias;
  }
}

// ---------------------------------------------------------------------------
// 2a) Partial Gram: Gp[split,b] += h_b[s0:s0+512]^T @ h_b[s0:s0+512]
// wave = 64x64 block of G (4x4 tiles, 16 accumulators).
// grid: (4, 4, B*GSPLIT), block: 32.  z: split = z/BATCH, b = z%BATCH
// ---------------------------------------------------------------------------
__global__ void k_gram_part(const float* __restrict__ h, float* __restrict__ gpart) {
  int lane  = threadIdx.x & 31;
  int n0    = blockIdx.x * 64;
  int m0    = blockIdx.y * 64;
  int z     = blockIdx.z;
  int b     = z % BATCH;
  int split = z / BATCH;
  const float* hb = h + (size_t)b * SEQ * D_MODEL;
  int s0 = split * (SEQ / GSPLIT);
  int s1 = s0 + (SEQ / GSPLIT);

  v8f acc[4][4] = {};
#pragma unroll 2
  for (int k0 = s0; k0 < s1; k0 += 4) {
    v2f av[4], bv[4];
#pragma unroll
    for (int i = 0; i < 4; ++i) av[i] = load_a_tr(hb, D_MODEL, m0 + 16 * i, k0, lane);
#pragma unroll
    for (int j = 0; j < 4; ++j) bv[j] = load_b_rm(hb, D_MODEL, n0 + 16 * j, k0, lane);
#pragma unroll
    for (int i = 0; i < 4; ++i)
#pragma unroll
      for (int j = 0; j < 4; ++j)
        acc[i][j] = wmma_f32(av[i], bv[j], acc[i][j]);
  }
  float* Gp = gpart + (size_t)z * D_MODEL * D_MODEL;
#pragma unroll
  for (int i = 0; i < 4; ++i)
#pragma unroll
    for (int j = 0; j < 4; ++j)
      store_d(Gp, D_MODEL, m0 + 16 * i, n0 + 16 * j, lane, acc[i][j]);
}

// 2b) G[b] = sum over splits of Gp[split,b]   (elementwise)
// grid: B*65536/256 blocks, block 256
__global__ void k_greduce(const float* __restrict__ gpart, float* __restrict__ G) {
  size_t i = (size_t)blockIdx.x * blockDim.x + threadIdx.x;
  const size_t GN = (size_t)BATCH * D_MODEL * D_MODEL;
  float acc = 0.f;
#pragma unroll
  for (int c = 0; c < GSPLIT; ++c) acc += gpart[(size_t)c * GN + i];
  G[i] = acc;
}

// ---------------------------------------------------------------------------
// 3) tiny batched 256^3 GEMM:  D[b] = opA(A[b]) @ B[b]; stride 0 = shared.
// grid: (16, 16, B), block: 32
// ---------------------------------------------------------------------------
template <bool TRANSA>
__global__ void k_gemm256(const float* __restrict__ A, const float* __restrict__ Bm,
                          float* __restrict__ Dm, long aStride, long bStride) {
  int lane = threadIdx.x & 31;
  int n0 = blockIdx.x * 16;
  int m0 = blockIdx.y * 16;
  int b  = blockIdx.z;
  const float* Ab = A + (size_t)b * aStride;
  const float* Bb = Bm + (size_t)b * bStride;
  v8f acc = {};
#pragma unroll 4
  for (int k0 = 0; k0 < D_MODEL; k0 += 4) {
    v2f a  = TRANSA ? load_a_tr(Ab, D_MODEL, m0, k0, lane)
                    : load_a_rm(Ab, D_MODEL, m0, k0, lane);
    v2f bb = load_b_rm(Bb, D_MODEL, n0, k0, lane);
    acc = wmma_f32(a, bb, acc);
  }
  store_d(Dm + (size_t)b * D_MODEL * D_MODEL, D_MODEL, m0, n0, lane, acc);
}

// ---------------------------------------------------------------------------
// 4) h_out = h_in + RES_SCALE * (h_in @ Wt[b]); wave = 16 rows x 64 cols
// grid: (4, 256, B), block: 32
// ---------------------------------------------------------------------------
__global__ void k_update(const float* __restrict__ hin, const float* __restrict__ Wt,
                         float* __restrict__ hout) {
  int lane = threadIdx.x & 31;
  int n0 = blockIdx.x * 64;
  int m0 = blockIdx.y * 16;
  int b  = blockIdx.z;
  const float* hb = hin + (size_t)b * SEQ * D_MODEL;
  const float* Wb = Wt + (size_t)b * D_MODEL * D_MODEL;
  float*       ho = hout + (size_t)b * SEQ * D_MODEL;
  v8f acc[4] = {};
#pragma unroll 2
  for (int k0 = 0; k0 < D_MODEL; k0 += 4) {
    v2f a = load_a_rm(hb, D_MODEL, m0, k0, lane);
#pragma unroll
    for (int j = 0; j < 4; ++j) {
      v2f bb = load_b_rm(Wb, D_MODEL, n0 + 16 * j, k0, lane);
      acc[j] = wmma_f32(a, bb, acc[j]);
    }
  }
  int colb  = lane & 15;
  int rbase = m0 + ((lane >> 4) << 3);
#pragma unroll
  for (int j = 0; j < 4; ++j)
#pragma unroll
    for (int r = 0; r < 8; ++r) {
      size_t idx = (size_t)(rbase + r) * D_MODEL + n0 + 16 * j + colb;
      ho[idx] = hb[idx] + RES_SCALE * acc[j][r];
    }
}

// ---------------------------------------------------------------------------
// 5a) partial column sums: cpart[(b*CSPLIT+c)*256+t] = sum over chunk rows
// grid: (CSPLIT, B), block: 256
// ---------------------------------------------------------------------------
__global__ void k_colsum(const float* __restrict__ h, float* __restrict__ cpart) {
  int c = blockIdx.x, b = blockIdx.y, t = threadIdx.x;
  const int rows = SEQ / CSPLIT;  // 128
  const float* hb = h + (size_t)b * SEQ * D_MODEL + (size_t)c * rows * D_MODEL;
  float acc = 0.f;
  for (int s = 0; s < rows; ++s) acc += hb[(size_t)s * D_MODEL + t];
  cpart[((size_t)b * CSPLIT + c) * D_MODEL + t] = acc;
}

// 5b) out[b] = colsum @ W_out + b_out; LDS tree reduction
// grid: B, block: 256
__global__ void k_final(const float* __restrict__ cpart, const float* __restrict__ Wout,
                        const float* __restrict__ bout, float* __restrict__ out) {
  __shared__ float red[D_MODEL];
  int b = blockIdx.x, t = threadIdx.x;
  float acc = 0.f;
#pragma unroll
  for (int c = 0; c < CSPLIT; ++c)
    acc += cpart[((size_t)b * CSPLIT + c) * D_MODEL + t];
  red[t] = acc * Wout[t];
  __syncthreads();
  for (int off = 128; off > 0; off >>= 1) {
    if (t < off) red[t] += red[t + off];
    __syncthreads();
  }
  if (t == 0) out[b] = red[0] + bout[0];
}

// ---------------------------------------------------------------------------
extern "C" void kernel_launch(void* const* d_in, const int* in_sizes, int n_in,
                              void* d_out, int out_size, void* d_ws, size_t ws_size,
                              hipStream_t stream) {
  const float* x  = (const float*)d_in[0];
  const float* We = (const float*)d_in[1];
  const float* be = (const float*)d_in[2];
  const float* Wq = (const float*)d_in[3];  // [2,256,256]
  const float* Wk = (const float*)d_in[4];
  const float* Wv = (const float*)d_in[5];
  const float* Wo = (const float*)d_in[6];
  const float* bo = (const float*)d_in[7];
  float* out = (float*)d_out;
  float* ws  = (float*)d_ws;

  const size_t HN = (size_t)BATCH * SEQ * D_MODEL;      // 4,194,304 floats
  const size_t GN = (size_t)BATCH * D_MODEL * D_MODEL;  //   262,144 floats
  float* hA    = ws;
  float* hB    = hA + HN;
  float* bufA  = hB + HN;                 // G, then T2
  float* bufB  = bufA + GN;               // T1, then W~
  float* gpart = bufB + GN;               // GSPLIT * GN
  float* cpart = gpart + (size_t)GSPLIT * GN;  // B*CSPLIT*256
  const long WSTRIDE = (long)D_MODEL * D_MODEL;

  dim3 wave(32);

  // h0 = x @ We + be
  k_embed<<<dim3(4, (BATCH * SEQ) / 16), wave, 0, stream>>>(x, We, be, hA);

  const float* hcur = hA;
  float*       hnxt = hB;
  for (int L = 0; L < 2; ++L) {
    const float* wq = Wq + (size_t)L * WSTRIDE;
    const float* wk = Wk + (size_t)L * WSTRIDE;
    const float* wv = Wv + (size_t)L * WSTRIDE;

    // G = h^T h  (split-S partials, then reduce)
    k_gram_part<<<dim3(4, 4, BATCH * GSPLIT), wave, 0, stream>>>(hcur, gpart);
    k_greduce<<<dim3((unsigned)(GN / 256)), dim3(256), 0, stream>>>(gpart, bufA);
    // T1 = G @ Wv
    k_gemm256<false><<<dim3(16, 16, BATCH), wave, 0, stream>>>(bufA, wv, bufB, WSTRIDE, 0L);
    // T2 = Wk^T @ T1
    k_gemm256<true><<<dim3(16, 16, BATCH), wave, 0, stream>>>(wk, bufB, bufA, 0L, WSTRIDE);
    // W~ = Wq @ T2
    k_gemm256<false><<<dim3(16, 16, BATCH), wave, 0, stream>>>(wq, bufA, bufB, 0L, WSTRIDE);
    // h_next = h + 0.1 * h @ W~
    k_update<<<dim3(4, SEQ / 16, BATCH), wave, 0, stream>>>(hcur, bufB, hnxt);

    float* t = (float*)hcur; hcur = hnxt; hnxt = t;
  }

  // out = (sum_s h) @ W_out + b_out
  k_colsum<<<dim3(CSPLIT, BATCH), dim3(256), 0, stream>>>(hcur, cpart);
  k_final<<<dim3(BATCH), dim3(256), 0, stream>>>(cpart, Wo, bo, out);
}